// SemiCurrSinkhornKnopp_69200513073733
// MI455X (gfx1250) — compile-verified
//
#include <hip/hip_runtime.h>
#include <hip/hip_bf16.h>
#include <math.h>
#include <stdint.h>

#define KC    1024      // K
#define NR    32768     // N
#define TMB   128       // rows per block in iteration kernel (8 waves x 16 rows)
#define THREADS_A 256

typedef __attribute__((ext_vector_type(2))) float v2f;
typedef __attribute__((ext_vector_type(8))) float v8f;
typedef __attribute__((ext_vector_type(4))) int   v4i;

#if defined(__has_builtin)
#  if __has_builtin(__builtin_amdgcn_wmma_f32_16x16x4_f32)
#    define HAVE_WMMA_F32X4 1
#  endif
#  if __has_builtin(__builtin_amdgcn_global_load_async_to_lds_b128)
#    define HAVE_ASYNC_B128 1
#  endif
#  if __has_builtin(__builtin_amdgcn_s_wait_asynccnt)
#    define HAVE_WAIT_ASYNC 1
#  endif
#endif
#ifndef HAVE_WMMA_F32X4
#  define HAVE_WMMA_F32X4 0
#endif
#ifndef HAVE_ASYNC_B128
#  define HAVE_ASYNC_B128 0
#endif
#ifndef HAVE_WAIT_ASYNC
#  define HAVE_WAIT_ASYNC 0
#endif

// ---------------------------------------------------------------------------
// Row stats: rowoff = 10*max_j(logit), mrow = 10*(max - logsumexp)  (<= 0)
// q'_ij = exp(10*x_ij - rowoff_i)  in (0,1];   Q_ij = e^{mrow_i} * q'_ij
// One wave per row, wave32 shuffles.
// ---------------------------------------------------------------------------
__global__ __launch_bounds__(256) void sk_rowstats(const float* __restrict__ logits,
                                                   float* __restrict__ rowoff,
                                                   float* __restrict__ mrow)
{
    const int lane = threadIdx.x & 31;
    const int wave = threadIdx.x >> 5;
    const int r = blockIdx.x * 8 + wave;
    const float* lr = logits + (size_t)r * KC;

    float mx = -3.4e38f;
    for (int j = lane; j < KC; j += 32) mx = fmaxf(mx, lr[j]);
    for (int o = 16; o; o >>= 1) mx = fmaxf(mx, __shfl_xor(mx, o, 32));
    float s = 0.f;
    for (int j = lane; j < KC; j += 32) s += __expf(lr[j] - mx);
    for (int o = 16; o; o >>= 1) s += __shfl_xor(s, o, 32);
    if (lane == 0) {
        const float lse = mx + logf(s);
        rowoff[r] = 10.f * mx;
        mrow[r]   = 10.f * (mx - lse);
    }
}

// ---------------------------------------------------------------------------
// Init: b0 = 1/(K+1) everywhere; bf = b0/exp(log max b0) = 1; t = 0; done = 0
// ---------------------------------------------------------------------------
__global__ __launch_bounds__(1024) void sk_init(double* __restrict__ bvec,
                                                float* __restrict__ bf,
                                                double* __restrict__ tvec,
                                                double* __restrict__ bslog,
                                                int* __restrict__ done)
{
    const int tid = threadIdx.x;
    const double b0 = 1.0 / 1025.0;
    bvec[tid] = b0;
    bf[tid]   = 1.0f;
    tvec[tid] = 0.0;
    if (tid == 0) {
        bvec[KC] = b0;
        tvec[KC] = 0.0;
        bslog[0] = log(b0);
        done[0]  = 0;
    }
}

// ---------------------------------------------------------------------------
// Fused iteration kernel: one pass over logits (L2-resident).
//   pass 1 (WMMA f32 16x16x4): dotp_i = sum_j q'_ij * bf_j
//           a_i  = (1/N) / (e^{lbs}*dotp + e^{-m_i}*b_K)
//           a'_i = a_i * e^{m_i}            (stored; plan/t use a')
//   pass 2: t_j += sum_i q'_ij * a'_i       (f64 global atomics)
//           t_K += sum_i a_i                (dustbin column of ones)
// ---------------------------------------------------------------------------
__global__ __launch_bounds__(THREADS_A) void sk_iter_rows(
    const float*  __restrict__ logits,
    const float*  __restrict__ rowoff,
    const float*  __restrict__ mrow,
    const double* __restrict__ bvec,
    const float*  __restrict__ bf,
    const double* __restrict__ bslog,
    double*       __restrict__ aprime,
    double*       __restrict__ tvec,
    const int*    __restrict__ done)
{
    if (done[0]) return;   // uniform: EXEC stays all-ones for WMMA path

    __shared__ float  sBf[KC];
    __shared__ double sAp[TMB];
    __shared__ double sDust;

    const int tid  = threadIdx.x;
    const int lane = tid & 31;
    const int wave = tid >> 5;
    const int r0b  = blockIdx.x * TMB;

    if (tid == 0) sDust = 0.0;

    // Stage scaled-b (f32, 4KB) into LDS: 256 threads x one async b128 each.
#if HAVE_ASYNC_B128
    {
        const float* src = bf + tid * 4;
        __builtin_amdgcn_global_load_async_to_lds_b128(
            (__attribute__((address_space(1))) v4i*)(uintptr_t)src,
            (__attribute__((address_space(3))) v4i*)(uint32_t)(uintptr_t)(&sBf[tid * 4]),
            0, 0);
#if HAVE_WAIT_ASYNC
        __builtin_amdgcn_s_wait_asynccnt(0);
#else
        asm volatile("s_wait_asynccnt 0" ::: "memory");
#endif
    }
#else
    ((float4*)sBf)[tid] = ((const float4*)bf)[tid];
#endif
    __syncthreads();

    // ---- pass 1: WMMA dot. A = bf slice replicated over M (documented 16x4
    // f32 layout: vgpr0 = K0|K2 per lane half, vgpr1 = K1|K3). B = q' tile,
    // 16 rows (N = lane&15) x 4 K. C(m,n) = sum_k bf[jb+k]*q'[row n][jb+k],
    // replicated over m -> each lane reads its row's dot from c[0].
    const int rw     = r0b + wave * 16;
    const int myrow  = rw + (lane & 15);
    const int colSel = (lane >> 4) << 1;     // 0 for lanes 0-15, 2 for 16-31
    const float off  = rowoff[myrow];
    const float2* lrow = (const float2*)(logits + (size_t)myrow * KC);

#if HAVE_WMMA_F32X4
    v8f c0 = {0.f,0.f,0.f,0.f,0.f,0.f,0.f,0.f};
    v8f c1 = {0.f,0.f,0.f,0.f,0.f,0.f,0.f,0.f};
#else
    float dAcc = 0.f;
#endif

    for (int jb = 0; jb < KC; jb += 8) {
        const int j0 = jb + colSel;
        const int j1 = jb + 4 + colSel;
        const float2 x0 = lrow[j0 >> 1];
        const float2 x1 = lrow[j1 >> 1];
        v2f q0, q1, b0, b1;
        q0.x = __expf(fmaf(10.f, x0.x, -off));
        q0.y = __expf(fmaf(10.f, x0.y, -off));
        q1.x = __expf(fmaf(10.f, x1.x, -off));
        q1.y = __expf(fmaf(10.f, x1.y, -off));
        b0.x = sBf[j0]; b0.y = sBf[j0 + 1];
        b1.x = sBf[j1]; b1.y = sBf[j1 + 1];
#if HAVE_WMMA_F32X4
        c0 = __builtin_amdgcn_wmma_f32_16x16x4_f32(false, b0, false, q0,
                                                   (short)0, c0, false, false);
        c1 = __builtin_amdgcn_wmma_f32_16x16x4_f32(false, b1, false, q1,
                                                   (short)0, c1, false, false);
#else
        dAcc = fmaf(q0.x, b0.x, dAcc); dAcc = fmaf(q0.y, b0.y, dAcc);
        dAcc = fmaf(q1.x, b1.x, dAcc); dAcc = fmaf(q1.y, b1.y, dAcc);
#endif
    }

#if HAVE_WMMA_F32X4
    const float dotp = c0[0] + c1[0];             // row (lane&15)'s dot
#else
    const float dotp = dAcc + __shfl_xor(dAcc, 16, 32);
#endif

    const double lbs = bslog[0];                  // max_j log b  (scalar)
    const double es  = exp(lbs);
    const double bK  = bvec[KC];                  // dustbin b (f64, exact)

    if (lane < 16) {
        const int r     = rw + lane;
        const double m  = (double)mrow[r];        // in [-69.3, 0]
        const double em = exp(m);
        const double den = es * (double)dotp + bK / em;   // e^{-m} * b_K
        const double a   = (1.0 / (double)NR) / den;      // plain a_i
        const double ap  = a * em;                        // a' = a * e^m
        aprime[r] = ap;
        sAp[wave * 16 + lane] = ap;
        atomicAdd(&sDust, a);                     // ds_add_f64 (dustbin col)
    }
    __syncthreads();

    // ---- pass 2: t_j partials (f64), coalesced re-read of logits (L2-hot).
    double acc0 = 0.0, acc1 = 0.0, acc2 = 0.0, acc3 = 0.0;
    const int j0c = tid, j1c = tid + 256, j2c = tid + 512, j3c = tid + 768;
    for (int i = 0; i < TMB; ++i) {
        const int r = r0b + i;
        const float* lr = logits + (size_t)r * KC;
        if (i + 1 < TMB)
            __builtin_prefetch(logits + (size_t)(r + 1) * KC + tid, 0, 1);
        const float offi = rowoff[r];             // uniform -> scalar load
        const double ai  = sAp[i];
        acc0 += (double)__expf(fmaf(10.f, lr[j0c], -offi)) * ai;
        acc1 += (double)__expf(fmaf(10.f, lr[j1c], -offi)) * ai;
        acc2 += (double)__expf(fmaf(10.f, lr[j2c], -offi)) * ai;
        acc3 += (double)__expf(fmaf(10.f, lr[j3c], -offi)) * ai;
    }
    atomicAdd(&tvec[j0c], acc0);                  // global_atomic_add_f64
    atomicAdd(&tvec[j1c], acc1);
    atomicAdd(&tvec[j2c], acc2);
    atomicAdd(&tvec[j3c], acc3);
    if (tid == 0) atomicAdd(&tvec[KC], sDust);
}

// ---------------------------------------------------------------------------
// b update (single block, K+1 entries): b = Pb/t, power-damp non-dustbin,
// err = ||b - b_old||, dynamic f32 rescale of b for next WMMA pass,
// reset t, set done flag on convergence.
// ---------------------------------------------------------------------------
__global__ __launch_bounds__(1024) void sk_update_b(double* __restrict__ bvec,
                                                    float*  __restrict__ bf,
                                                    double* __restrict__ tvec,
                                                    double* __restrict__ bslog,
                                                    int*    __restrict__ done)
{
    if (done[0]) return;
    __shared__ double sSum[32];
    __shared__ double sMax[32];
    const int tid = threadIdx.x;
    const double fi = 1.0 / 1.1;                  // GAMMA/(GAMMA+EPSILON)
    const double PbReg = 0.5 / 1024.0;            // RHO/K

    double t = fmax(tvec[tid], 1e-290);
    const double bo  = bvec[tid];
    const double bn  = pow(PbReg / t, fi);        // power damping, j < K
    const double lbj = log(bn);
    double e2 = (bn - bo) * (bn - bo);
    if (tid == 0) {
        const double tK  = fmax(tvec[KC], 1e-290);
        const double bKn = 0.5 / tK;              // dustbin: no power
        const double dK  = bKn - bvec[KC];
        e2 += dK * dK;
        bvec[KC] = bKn;
        tvec[KC] = 0.0;
    }
    bvec[tid] = bn;
    tvec[tid] = 0.0;

    double rs = e2, rm = lbj;
    for (int o = 16; o; o >>= 1) {
        rs += __shfl_xor(rs, o, 32);
        rm  = fmax(rm, __shfl_xor(rm, o, 32));
    }
    const int wv = tid >> 5, ln = tid & 31;
    if (ln == 0) { sSum[wv] = rs; sMax[wv] = rm; }
    __syncthreads();
    if (wv == 0) {
        double s2 = sSum[ln], m2 = sMax[ln];
        for (int o = 16; o; o >>= 1) {
            s2 += __shfl_xor(s2, o, 32);
            m2  = fmax(m2, __shfl_xor(m2, o, 32));
        }
        if (ln == 0) {
            sMax[0]  = m2;
            bslog[0] = m2;
            if (sqrt(s2) <= 1e-6) done[0] = 1;
        }
    }
    __syncthreads();
    bf[tid] = (float)exp(lbj - sMax[0]);          // scaled f32 b, <= 1
}

// ---------------------------------------------------------------------------
// plan_ij = n * a_i * Q_ij * b_j = n * a'_i * q'_ij * b_j   (f32 out, no dustbin)
// ---------------------------------------------------------------------------
__global__ __launch_bounds__(256) void sk_plan(const float*  __restrict__ logits,
                                               const float*  __restrict__ rowoff,
                                               const double* __restrict__ aprime,
                                               const double* __restrict__ bvec,
                                               float* __restrict__ out)
{
    const int r = blockIdx.x;
    const float off  = rowoff[r];
    const double apn = aprime[r] * (double)NR;
    const float* lr  = logits + (size_t)r * KC;
    float* orow      = out + (size_t)r * KC;
#pragma unroll
    for (int s = 0; s < 4; ++s) {
        const int j = threadIdx.x + s * 256;
        const double q = (double)__expf(fmaf(10.f, lr[j], -off));
        orow[j] = (float)(apn * q * bvec[j]);
    }
}

// ---------------------------------------------------------------------------
extern "C" void kernel_launch(void* const* d_in, const int* in_sizes, int n_in,
                              void* d_out, int out_size, void* d_ws, size_t ws_size,
                              hipStream_t stream)
{
    (void)in_sizes; (void)n_in; (void)out_size; (void)ws_size;
    const float* logits = (const float*)d_in[0];
    float* out = (float*)d_out;

    char* w = (char*)d_ws;
    auto take = [&](size_t bytes) -> char* {
        char* p = w;
        w += (bytes + 255) & ~(size_t)255;
        return p;
    };
    double* aprime = (double*)take((size_t)NR * 8);       // a' (f64)
    double* bvec   = (double*)take((KC + 1) * 8);         // b  (f64)
    double* tvec   = (double*)take((KC + 1) * 8);         // Q^T a accum (f64)
    double* bslog  = (double*)take(8);                    // max_j log b
    int*    done   = (int*)take(8);                       // convergence flag
    float*  rowoff = (float*)take((size_t)NR * 4);        // 10*rowmax
    float*  mrow   = (float*)take((size_t)NR * 4);        // 10*(rowmax - lse)
    float*  bf     = (float*)take((size_t)KC * 4);        // scaled b (f32)

    sk_rowstats<<<NR / 8, 256, 0, stream>>>(logits, rowoff, mrow);
    sk_init<<<1, KC, 0, stream>>>(bvec, bf, tvec, bslog, done);
    for (int it = 0; it < 1000; ++it) {   // NUM_ITERMAX; done-flag no-ops rest
        sk_iter_rows<<<NR / TMB, THREADS_A, 0, stream>>>(
            logits, rowoff, mrow, bvec, bf, bslog, aprime, tvec, done);
        sk_update_b<<<1, KC, 0, stream>>>(bvec, bf, tvec, bslog, done);
    }
    sk_plan<<<NR, 256, 0, stream>>>(logits, rowoff, aprime, bvec, out);
}